// IDetector3D_28776280883737
// MI455X (gfx1250) — compile-verified
//
#include <hip/hip_runtime.h>
#include <hip/hip_bf16.h>
#include <math.h>

#define B_     4
#define N_     16384
#define TOPK_  1024
#define POSTK_ 256
#define MGT_   50
#define RROI_  512
#define NMS_TH 0.1f

typedef float v2f __attribute__((ext_vector_type(2)));
typedef float v8f __attribute__((ext_vector_type(8)));

__device__ inline unsigned ballot32(bool p) {
#if __has_builtin(__builtin_amdgcn_ballot_w32)
    return __builtin_amdgcn_ballot_w32(p);
#else
    return (unsigned)__ballot(p);
#endif
}

__device__ inline void boxCorners(float x, float y, float dx, float dy, float h,
                                  float* cx, float* cy) {
    float c = cosf(h), s = sinf(h);
    const float lxs[4] = {0.5f, -0.5f, -0.5f, 0.5f};
    const float lys[4] = {0.5f, 0.5f, -0.5f, -0.5f};
#pragma unroll
    for (int k = 0; k < 4; k++) {
        float lx = lxs[k] * dx, ly = lys[k] * dy;
        cx[k] = x + lx * c - ly * s;
        cy[k] = y + lx * s + ly * c;
    }
}

// Sutherland-Hodgman clip of quad A by convex quad B, returns intersection area.
// Mirrors reference: intersection point emitted before current point, d>=0 inside,
// denom guard 1e-12, area = max(0.5*shoelace, 0).
__device__ float polyClipArea(const float* ax, const float* ay,
                              const float* bx, const float* by) {
    float px[8], py[8];
    int n = 4;
#pragma unroll
    for (int k = 0; k < 4; k++) { px[k] = ax[k]; py[k] = ay[k]; }
#pragma unroll
    for (int k = 4; k < 8; k++) { px[k] = 0.f; py[k] = 0.f; }
    for (int k = 0; k < 4; k++) {
        float a0 = bx[k], a1 = by[k];
        int k2 = (k + 1) & 3;
        float ex = bx[k2] - a0, ey = by[k2] - a1;
        float qx[8], qy[8];
        int m = 0;
        for (int i = 0; i < 8; i++) {
            if (i >= n) break;
            int ip = (i == 0) ? (n - 1) : (i - 1);
            float dc = ex * (py[i] - a1) - ey * (px[i] - a0);
            float dp = ex * (py[ip] - a1) - ey * (px[ip] - a0);
            bool in_c = dc >= 0.f, in_p = dp >= 0.f;
            if (in_c != in_p) {
                float den = dp - dc;
                float t = dp / (fabsf(den) < 1e-12f ? 1e-12f : den);
                if (m < 8) {
                    qx[m] = px[ip] + t * (px[i] - px[ip]);
                    qy[m] = py[ip] + t * (py[i] - py[ip]);
                    m++;
                }
            }
            if (in_c && m < 8) { qx[m] = px[i]; qy[m] = py[i]; m++; }
        }
        n = m;
        for (int i = 0; i < 8; i++) {
            px[i] = (i < m) ? qx[i] : 0.f;
            py[i] = (i < m) ? qy[i] : 0.f;
        }
    }
    float s = 0.f;
    for (int i = 0; i < 8; i++) {
        if (i >= n) break;
        int nx = (i + 1 == n) ? 0 : (i + 1);
        s += px[i] * py[nx] - px[nx] * py[i];
    }
    return fmaxf(0.5f * s, 0.f);
}

// ---------------- K0: zero outputs + max-iou scratch ----------------
__global__ void k_zero(float* out, int nOut, float* mp, float* mr, int nM) {
    int i = blockIdx.x * blockDim.x + threadIdx.x;
    if (i < nOut) out[i] = 0.f;
    if (i < nM) { mp[i] = 0.f; mr[i] = 0.f; }
}

// ---------------- K1: sigmoid(max logit), argmax label ----------------
__global__ void k_scores(const float* __restrict__ logits, float* scores, int* labels) {
    int i = blockIdx.x * blockDim.x + threadIdx.x;
    if (i >= B_ * N_) return;
    float l0 = logits[i * 3 + 0], l1 = logits[i * 3 + 1], l2 = logits[i * 3 + 2];
    int lab = 0; float mx = l0;
    if (l1 > mx) { mx = l1; lab = 1; }
    if (l2 > mx) { mx = l2; lab = 2; }
    scores[i] = 1.f / (1.f + expf(-mx));
    labels[i] = lab + 1;
}

// ------------- K2: full bitonic sort in 128KB dynamic LDS (one WG/sample) -------------
__global__ void k_sort(const float* __restrict__ scores, int* topIdx, float* topScore) {
    extern __shared__ unsigned long long key[];
    int b = blockIdx.x, t = threadIdx.x;
    for (int i = t; i < N_; i += blockDim.x) {
        unsigned sb = __float_as_uint(scores[b * N_ + i]); // sigmoid>0 -> monotone bits
        key[i] = ((unsigned long long)sb << 32) | (unsigned)(N_ - 1 - i); // ties: low idx first
    }
    __syncthreads();
    for (int k = 2; k <= N_; k <<= 1) {
        for (int j = k >> 1; j > 0; j >>= 1) {
            for (int i = t; i < N_; i += blockDim.x) {
                int l = i ^ j;
                if (l > i) {
                    unsigned long long a = key[i], c = key[l];
                    bool desc = ((i & k) == 0);
                    bool sw = desc ? (a < c) : (a > c);
                    if (sw) { key[i] = c; key[l] = a; }
                }
            }
            __syncthreads();
        }
    }
    if (t < TOPK_) {
        unsigned long long kk = key[t];
        topScore[b * TOPK_ + t] = __uint_as_float((unsigned)(kk >> 32));
        topIdx[b * TOPK_ + t] = (N_ - 1) - (int)(kk & 0xFFFFFFFFull);
    }
}

// ------------- K3: gather top boxes, precompute corners/area/radius/center -------------
__global__ void k_prep(const float* __restrict__ preds, const int* __restrict__ topIdx,
                       const int* __restrict__ labels, float* tb7, float* cxy,
                       float* area, float* rad, float* ctr, int* tlab) {
    int i = blockIdx.x * blockDim.x + threadIdx.x;
    if (i >= B_ * TOPK_) return;
    int b = i / TOPK_;
    int idx = topIdx[i];
    const float* bx = preds + (size_t)(b * N_ + idx) * 7;
    float x = bx[0], y = bx[1], dx = bx[3], dy = bx[4], h = bx[6];
#pragma unroll
    for (int c = 0; c < 7; c++) tb7[i * 7 + c] = bx[c];
    float cx[4], cy[4];
    boxCorners(x, y, dx, dy, h, cx, cy);
#pragma unroll
    for (int k = 0; k < 4; k++) { cxy[i * 8 + k] = cx[k]; cxy[i * 8 + 4 + k] = cy[k]; }
    area[i] = dx * dy;
    rad[i] = 0.5f * sqrtf(dx * dx + dy * dy);
    ctr[i * 2 + 0] = x; ctr[i * 2 + 1] = y;
    tlab[i] = labels[b * N_ + idx];
}

// ------------- K4: WMMA distance prefilter + clip -> suppression bitmask -------------
// One wave computes a 16x16 tile of the pair matrix. V_WMMA_F32_16X16X4_F32 gives
// D[m][n] = x_i*x_j + y_i*y_j (K=0,1 carry x,y; lanes 16-31 carry K=2,3 = 0).
__global__ void k_mask(const float* __restrict__ ctr, const float* __restrict__ cxy,
                       const float* __restrict__ area, const float* __restrict__ rad,
                       unsigned short* __restrict__ mask16) {
    int wave = blockIdx.x * (blockDim.x >> 5) + (threadIdx.x >> 5);
    int lane = threadIdx.x & 31;
    int b = wave >> 12;            // 4096 tiles per batch
    int t = wave & 4095;
    int ti = t >> 6, tj = t & 63;
    int bi = ti * 16, bj = tj * 16;
    const float* ctrb = ctr + b * TOPK_ * 2;

    v2f a, bm;
    if (lane < 16) {
        a.x = ctrb[(bi + lane) * 2 + 0]; a.y = ctrb[(bi + lane) * 2 + 1];
        bm.x = ctrb[(bj + lane) * 2 + 0]; bm.y = ctrb[(bj + lane) * 2 + 1];
    } else { a.x = 0.f; a.y = 0.f; bm.x = 0.f; bm.y = 0.f; }
    v8f c;
#pragma unroll
    for (int q = 0; q < 8; q++) c[q] = 0.f;
    v8f d = __builtin_amdgcn_wmma_f32_16x16x4_f32(false, a, false, bm, (short)0, c,
                                                  false, false);

    int n = lane & 15;
    int j = bj + n;
    float jx = ctrb[j * 2], jy = ctrb[j * 2 + 1];
    float nj = jx * jx + jy * jy;
    float rj = rad[b * TOPK_ + j];
    float aj = area[b * TOPK_ + j];
    float cbx[4], cby[4];
#pragma unroll
    for (int k2 = 0; k2 < 4; k2++) {
        cbx[k2] = cxy[(size_t)(b * TOPK_ + j) * 8 + k2];
        cby[k2] = cxy[(size_t)(b * TOPK_ + j) * 8 + 4 + k2];
    }
#pragma unroll
    for (int r = 0; r < 8; r++) {
        int m = (lane < 16) ? r : (r + 8);   // C/D layout: VGPR r -> M=r (lanes 0-15), M=r+8 (16-31)
        int i = bi + m;
        float ix = ctrb[i * 2], iy = ctrb[i * 2 + 1];
        float ni = ix * ix + iy * iy;
        float dist2 = ni + nj - 2.f * d[r];
        float rr = rad[b * TOPK_ + i] + rj;
        float iou = 0.f;
        if (dist2 <= rr * rr + 1e-2f) {      // far pairs cannot intersect -> iou 0
            float cax[4], cay[4];
#pragma unroll
            for (int k2 = 0; k2 < 4; k2++) {
                cax[k2] = cxy[(size_t)(b * TOPK_ + i) * 8 + k2];
                cay[k2] = cxy[(size_t)(b * TOPK_ + i) * 8 + 4 + k2];
            }
            float inter = polyClipArea(cax, cay, cbx, cby);
            float ai = area[b * TOPK_ + i];
            iou = inter / fmaxf(ai + aj - inter, 1e-6f);
        }
        unsigned ball = ballot32(iou > NMS_TH);
        // bits 0-15: row r, cols bj..bj+15 ; bits 16-31: row r+8, same cols
        if (lane == 0)
            mask16[(size_t)(b * TOPK_ + bi + r) * 64 + tj] = (unsigned short)(ball & 0xFFFFu);
        if (lane == 16)
            mask16[(size_t)(b * TOPK_ + bi + r + 8) * 64 + tj] = (unsigned short)(ball >> 16);
    }
}

// ------------- K5: sequential NMS scan, one wave per sample -------------
__global__ void k_scan(const unsigned* __restrict__ mask, unsigned* keepw) {
    int b = blockIdx.x;
    int t = threadIdx.x;                    // 32 threads, word t = cols [32t,32t+32)
    unsigned kw = 0xFFFFFFFFu;
    const unsigned* mrow = mask + (size_t)b * TOPK_ * 32;
    for (int i = 0; i < TOPK_; i++) {
        unsigned wi = (unsigned)__shfl((int)kw, i >> 5, 32);
        if ((wi >> (i & 31)) & 1u) {
            unsigned m = mrow[(size_t)i * 32 + t];
            int iw = i >> 5;
            if (t < iw) m = 0u;
            else if (t == iw) {
                unsigned low = ((i & 31) == 31) ? 0xFFFFFFFFu : ((2u << (i & 31)) - 1u);
                m &= ~low;                  // only suppress j > i
            }
            kw &= ~m;
        }
    }
    keepw[b * 32 + t] = kw;
}

// ------------- K6: stable partition (== argsort(!keep)) and gather outputs -------------
__global__ void k_select(const unsigned* __restrict__ keepw, const float* __restrict__ tb7,
                         const float* __restrict__ tscore, const int* __restrict__ tlab,
                         float* fb, float* fs, float* fl, float* fvalid) {
    int b = blockIdx.x;
    int i = threadIdx.x;                    // 1024 threads
    __shared__ unsigned pre[33];
    if (i == 0) {
        unsigned s = 0; pre[0] = 0;
        for (int w = 0; w < 32; w++) { s += __popc(keepw[b * 32 + w]); pre[w + 1] = s; }
    }
    __syncthreads();
    unsigned kwv = keepw[b * 32 + (i >> 5)];
    bool kp = (kwv >> (i & 31)) & 1u;
    unsigned before = pre[i >> 5] + __popc(kwv & ((1u << (i & 31)) - 1u));
    unsigned nkeep = pre[32];
    int pos = kp ? (int)before : (int)(nkeep + ((unsigned)i - before));
    if (pos < POSTK_) {
        int ob = b * POSTK_ + pos;
#pragma unroll
        for (int c = 0; c < 7; c++)
            fb[ob * 7 + c] = kp ? tb7[(size_t)(b * TOPK_ + i) * 7 + c] : 0.f;
        fs[ob] = kp ? tscore[b * TOPK_ + i] : 0.f;
        fl[ob] = kp ? (float)tlab[b * TOPK_ + i] : 0.f;
        fvalid[ob] = kp ? 1.f : 0.f;
    }
}

// ------------- K7: iou3d(preds|rois, gts), atomicMax per gt -------------
__global__ void k_recall_pairs(const float* __restrict__ preds, const float* __restrict__ rois,
                               const float* __restrict__ gts, float* maxpred, float* maxroi) {
    const int PP = N_ + RROI_;
    int tid = blockIdx.x * blockDim.x + threadIdx.x;
    if (tid >= B_ * PP * MGT_) return;
    int b = tid / (PP * MGT_);
    int rem = tid % (PP * MGT_);
    int p = rem / MGT_;
    int g = rem % MGT_;
    const float* gb = gts + (size_t)(b * MGT_ + g) * 8;
    const float* ab = (p < N_) ? preds + (size_t)(b * N_ + p) * 7
                               : rois + (size_t)(b * RROI_ + (p - N_)) * 7;
    float rdx = ab[0] - gb[0], rdy = ab[1] - gb[1];
    float ra = 0.5f * sqrtf(ab[3] * ab[3] + ab[4] * ab[4]) +
               0.5f * sqrtf(gb[3] * gb[3] + gb[4] * gb[4]);
    float inter = 0.f;
    if (rdx * rdx + rdy * rdy <= ra * ra + 1e-2f) {
        float pax[4], pay[4], gax[4], gay[4];
        boxCorners(ab[0], ab[1], ab[3], ab[4], ab[6], pax, pay);
        boxCorners(gb[0], gb[1], gb[3], gb[4], gb[6], gax, gay);
        inter = polyClipArea(pax, pay, gax, gay);
    }
    float ahi = ab[2] + ab[5] * 0.5f, alo = ab[2] - ab[5] * 0.5f;
    float ghi = gb[2] + gb[5] * 0.5f, glo = gb[2] - gb[5] * 0.5f;
    float oh = fmaxf(fminf(ahi, ghi) - fmaxf(alo, glo), 0.f);
    float i3 = inter * oh;
    float va = ab[3] * ab[4] * ab[5], vb = gb[3] * gb[4] * gb[5];
    float iou = i3 / fmaxf(va + vb - i3, 1e-6f);
    float* dst = (p < N_) ? &maxpred[b * MGT_ + g] : &maxroi[b * MGT_ + g];
    atomicMax((unsigned*)dst, __float_as_uint(iou));   // iou >= 0 -> bit-monotone
}

// ------------- K8: gt_valid/last, recall counts, num_gt -------------
__global__ void k_finalize(const float* __restrict__ gts, const float* __restrict__ maxpred,
                           const float* __restrict__ maxroi, float* out_rec, float* out_ngt) {
    __shared__ int last[B_];
    int t = threadIdx.x;
    if (t < B_) {
        int l = -1;
        for (int g = 0; g < MGT_; g++) {
            float s = 0.f;
            for (int c = 0; c < 8; c++) s += gts[(size_t)(t * MGT_ + g) * 8 + c];
            if (s != 0.f) l = g;
        }
        last[t] = l;
        atomicAdd(out_ngt, (float)(l + 1));
    }
    __syncthreads();
    if (t < B_ * MGT_) {
        int b = t / MGT_, g = t % MGT_;
        if (g <= last[b]) {
            float mp = maxpred[b * MGT_ + g], mr = maxroi[b * MGT_ + g];
            const float th[3] = {0.3f, 0.5f, 0.7f};
#pragma unroll
            for (int k = 0; k < 3; k++) {
                if (mp > th[k]) atomicAdd(&out_rec[k], 1.f);
                if (mr > th[k]) atomicAdd(&out_rec[k], 1.f);
            }
        }
    }
}

extern "C" void kernel_launch(void* const* d_in, const int* in_sizes, int n_in,
                              void* d_out, int out_size, void* d_ws, size_t ws_size,
                              hipStream_t stream) {
    const float* preds  = (const float*)d_in[0];   // (4,16384,7)
    const float* logits = (const float*)d_in[1];   // (4,16384,3)
    const float* gts    = (const float*)d_in[2];   // (4,50,8)
    const float* rois   = (const float*)d_in[3];   // (4,512,7)
    (void)d_in; (void)in_sizes; (void)n_in; (void)ws_size;

    char* w = (char*)d_ws;
    float*          scores   = (float*)(w + 0);        // 262144 B
    int*            labels   = (int*)(w + 262144);     // 262144 B
    int*            topIdx   = (int*)(w + 524288);     // 16384 B
    float*          topScore = (float*)(w + 540672);   // 16384 B
    float*          tb7      = (float*)(w + 557056);   // 114688 B
    float*          cxy      = (float*)(w + 671744);   // 131072 B
    float*          area     = (float*)(w + 802816);   // 16384 B
    float*          rad      = (float*)(w + 819200);   // 16384 B
    float*          ctr      = (float*)(w + 835584);   // 32768 B
    int*            tlab     = (int*)(w + 868352);     // 16384 B
    unsigned short* mask16   = (unsigned short*)(w + 884736); // 524288 B
    unsigned*       keepw    = (unsigned*)(w + 1409024);      // 512 B
    float*          maxpred  = (float*)(w + 1409536);         // 800 B
    float*          maxroi   = (float*)(w + 1410560);         // 800 B

    float* out     = (float*)d_out;
    float* fb      = out;                 // 4*256*7
    float* fs      = out + 7168;          // 4*256
    float* fl      = out + 8192;          // 4*256
    float* fvalid  = out + 9216;          // 4*256
    float* out_rec = out + 10240;         // 3
    float* out_ngt = out + 10243;         // 1

    k_zero<<<(10244 + 255) / 256, 256, 0, stream>>>(out, 10244, maxpred, maxroi, B_ * MGT_);
    k_scores<<<(B_ * N_ + 255) / 256, 256, 0, stream>>>(logits, scores, labels);
    k_sort<<<B_, 1024, N_ * sizeof(unsigned long long), stream>>>(scores, topIdx, topScore);
    k_prep<<<(B_ * TOPK_ + 255) / 256, 256, 0, stream>>>(preds, topIdx, labels, tb7, cxy,
                                                         area, rad, ctr, tlab);
    // 16384 wave-tiles (4 batches x 64x64), 4 waves per 128-thread block
    k_mask<<<4096, 128, 0, stream>>>(ctr, cxy, area, rad, mask16);
    k_scan<<<B_, 32, 0, stream>>>((const unsigned*)mask16, keepw);
    k_select<<<B_, 1024, 0, stream>>>(keepw, tb7, topScore, tlab, fb, fs, fl, fvalid);
    {
        const int total = B_ * (N_ + RROI_) * MGT_;
        k_recall_pairs<<<(total + 255) / 256, 256, 0, stream>>>(preds, rois, gts,
                                                                maxpred, maxroi);
    }
    k_finalize<<<1, 256, 0, stream>>>(gts, maxpred, maxroi, out_rec, out_ngt);
}